// BiLSTM_CRF_73057393705184
// MI455X (gfx1250) — compile-verified
//
#include <hip/hip_runtime.h>
#include <hip/hip_bf16.h>
#include <math.h>

// Problem constants (match reference)
#define T_LEN 4096
#define EMB   50
#define HID   50
#define TAG   5
#define KPAD  52      // EMB padded to multiple of 4 for 16x16x4 WMMA
#define NFOUR 400     // 4*HID*2 directions, columns of fused input projection
#define START_TAG 3
#define STOP_TAG  4
#define NEG_INF  -10000.0f

typedef float v2f __attribute__((ext_vector_type(2)));
typedef float v8f __attribute__((ext_vector_type(8)));

__device__ __forceinline__ float sigmoidf_(float x) {
    return 1.0f / (1.0f + __expf(-x));
}

// ---------------------------------------------------------------------------
// Kernel 1a: embedding gather, padded to KPAD columns (zeros in cols 50..51)
// ---------------------------------------------------------------------------
__global__ void __launch_bounds__(64) gather_pad_kernel(
    const int* __restrict__ sentence,
    const float* __restrict__ embed,
    float* __restrict__ Xp)   // [T_LEN, KPAD]
{
    const int t = blockIdx.x;
    const int j = threadIdx.x;
    if (j < KPAD) {
        float v = 0.0f;
        if (j < EMB) {
            long long tok = (long long)sentence[t];
            v = embed[tok * EMB + j];
        }
        Xp[t * KPAD + j] = v;
    }
}

// ---------------------------------------------------------------------------
// Kernel 1b: pad combined input-projection weights & bias so the WMMA loop
// needs no guards:  Bpt[n][k] = Wcomb[n][k] (k<50) else 0,  n in [0,400)
// ---------------------------------------------------------------------------
__global__ void __launch_bounds__(64) pad_weights_kernel(
    const float* __restrict__ Wih_f,  // [200, 50]
    const float* __restrict__ Wih_b,  // [200, 50]
    const float* __restrict__ b_f,    // [200]
    const float* __restrict__ b_b,    // [200]
    float* __restrict__ Bpt,          // [NFOUR, KPAD]
    float* __restrict__ biasc)        // [NFOUR]
{
    const int n = blockIdx.x;         // 0..399
    const int j = threadIdx.x;
    const float* __restrict__ src =
        (n < 200) ? (Wih_f + n * EMB) : (Wih_b + (n - 200) * EMB);
    if (j < KPAD) {
        Bpt[n * KPAD + j] = (j < EMB) ? src[j] : 0.0f;
    }
    if (j == 0) {
        biasc[n] = (n < 200) ? b_f[n] : b_b[n - 200];
    }
}

// ---------------------------------------------------------------------------
// Kernel 2: fused input projection  Z = Xp @ Bpt^T + biasc
// fp32 WMMA 16x16x4, one wave per 16x16 tile.  M=4096, N=400, K=52 (13 steps).
// A frag (16x4 f32, ISA 7.12.2): lane L holds M=L&15, K = 4*kk + 2*(L>>4) + j
// B frag (4x16 f32, row-striped like C): lane L holds N=L&15, same K indexing
// C/D (16x16 f32): VGPR v, lane L -> row = v + 8*(L>>4), col = L&15
// Both fragment loads are clean 8-byte global_load_b64 (kbase is even, rows
// are 208 B): no guards, no exec divergence in the hot loop.
// ---------------------------------------------------------------------------
__global__ void __launch_bounds__(32) input_proj_wmma_kernel(
    const float* __restrict__ Xp,     // [T_LEN, KPAD]
    const float* __restrict__ Bpt,    // [NFOUR, KPAD] zero-padded
    const float* __restrict__ biasc,  // [NFOUR]
    float* __restrict__ Z)            // [T_LEN, NFOUR]
{
    const int lane = threadIdx.x;     // 0..31, wave32
    const int tm   = blockIdx.x;      // 0..255  (M tiles)
    const int tn   = blockIdx.y;      // 0..24   (N tiles, 25*16 == 400 exact)
    const int l15  = lane & 15;
    const int nh   = lane >> 4;       // half-wave select

    const int arow = tm * 16 + l15;           // A matrix row (time index)
    const int ncol = tn * 16 + l15;           // B/C matrix column (gate unit)

    const float* __restrict__ xa = Xp  + (long long)arow * KPAD;
    const float* __restrict__ bb = Bpt + (long long)ncol * KPAD;

    v8f acc = {};
#pragma unroll
    for (int kk = 0; kk < KPAD / 4; ++kk) {
        const int kbase = kk * 4 + 2 * nh;
        const v2f a = *(const v2f*)(xa + kbase);
        const v2f b = *(const v2f*)(bb + kbase);
        acc = __builtin_amdgcn_wmma_f32_16x16x4_f32(
            /*neg_a=*/false, a, /*neg_b=*/false, b,
            /*c_mod=*/(short)0, acc, /*reuse_a=*/false, /*reuse_b=*/false);
    }

    const float bias = biasc[ncol];
#pragma unroll
    for (int v = 0; v < 8; ++v) {
        const int orow = tm * 16 + v + 8 * nh;
        Z[(long long)orow * NFOUR + ncol] = acc[v] + bias;
    }
}

// ---------------------------------------------------------------------------
// Kernel 3: sequential LSTM recurrence, one workgroup per direction.
// Whh resident in LDS (40 KB of the 320 KB WGP pool); h in LDS; c in regs.
// ---------------------------------------------------------------------------
__global__ void __launch_bounds__(256) lstm_recurrence_kernel(
    const float* __restrict__ Z,      // [T_LEN, NFOUR] (bias + x-proj fused)
    const float* __restrict__ Whh_f,  // [200, 50]
    const float* __restrict__ Whh_b,  // [200, 50]
    const float* __restrict__ h0,     // [2, 50]
    const float* __restrict__ c0,     // [2, 50]
    float* __restrict__ hs_f,         // [T_LEN, HID]
    float* __restrict__ hs_b)         // [T_LEN, HID]
{
    __shared__ float whh[4 * HID * HID];   // 200*50
    __shared__ float h_lds[HID];
    __shared__ float a_lds[4 * HID];

    const int tid = threadIdx.x;
    const int dir = blockIdx.x;            // 0 = forward, 1 = backward
    const float* __restrict__ Whh = dir ? Whh_b : Whh_f;
    float* __restrict__ hs = dir ? hs_b : hs_f;

    for (int i = tid; i < 4 * HID * HID; i += 256) whh[i] = Whh[i];
    if (tid < HID) h_lds[tid] = h0[dir * HID + tid];
    float c_reg = (tid < HID) ? c0[dir * HID + tid] : 0.0f;
    __syncthreads();

    for (int s = 0; s < T_LEN; ++s) {
        const int t = dir ? (T_LEN - 1 - s) : s;
        if (tid < 4 * HID) {
            float acc = Z[(long long)t * NFOUR + dir * 200 + tid];
            const float* __restrict__ wr = &whh[tid * HID];
#pragma unroll 10
            for (int j = 0; j < HID; ++j) acc += wr[j] * h_lds[j];
            a_lds[tid] = acc;
        }
        __syncthreads();
        if (tid < HID) {
            const float ig = sigmoidf_(a_lds[tid]);
            const float fg = sigmoidf_(a_lds[HID + tid]);
            const float gg = tanhf(a_lds[2 * HID + tid]);
            const float og = sigmoidf_(a_lds[3 * HID + tid]);
            c_reg = fg * c_reg + ig * gg;
            const float hv = og * tanhf(c_reg);
            h_lds[tid] = hv;
            hs[(long long)t * HID + tid] = hv;
        }
        __syncthreads();
    }
}

// ---------------------------------------------------------------------------
// Kernel 4: tag projection + Viterbi forward scan + backtrace (single block).
// ---------------------------------------------------------------------------
__global__ void __launch_bounds__(256) feats_viterbi_kernel(
    const float* __restrict__ hs_f,   // [T_LEN, HID]
    const float* __restrict__ hs_b,   // [T_LEN, HID]
    const float* __restrict__ Wt,     // [TAG, 2*HID]
    const float* __restrict__ bt,     // [TAG]
    const float* __restrict__ trans,  // [TAG, TAG]
    float* __restrict__ feats,        // [T_LEN, TAG] scratch
    int*   __restrict__ bps,          // [T_LEN, TAG] scratch
    float* __restrict__ out)          // [T_LEN + 1]: path (as f32) + score
{
    __shared__ float tr[TAG * TAG];
    __shared__ float wt_s[TAG * 2 * HID];
    __shared__ float bt_s[TAG];
    __shared__ float fv[2][TAG];

    const int tid = threadIdx.x;
    if (tid < TAG * TAG) tr[tid] = trans[tid];
    for (int i = tid; i < TAG * 2 * HID; i += 256) wt_s[i] = Wt[i];
    if (tid < TAG) {
        bt_s[tid]  = bt[tid];
        fv[0][tid] = (tid == START_TAG) ? 0.0f : NEG_INF;
    }
    __syncthreads();

    // feats[t][tag] = hf[t].Wt[tag,0:50] + hb[t].Wt[tag,50:100] + bt[tag]
    for (int idx = tid; idx < T_LEN * TAG; idx += 256) {
        const int t = idx / TAG, tag = idx % TAG;
        const float* __restrict__ w = &wt_s[tag * 2 * HID];
        const float* __restrict__ hf = hs_f + (long long)t * HID;
        const float* __restrict__ hb = hs_b + (long long)t * HID;
        float acc = bt_s[tag];
#pragma unroll 10
        for (int j = 0; j < HID; ++j) acc += hf[j] * w[j];
#pragma unroll 10
        for (int j = 0; j < HID; ++j) acc += hb[j] * w[HID + j];
        feats[idx] = acc;
    }
    __syncthreads();

    // Viterbi forward: fv_new[next] = max_prev(fv[prev] + tr[next][prev]) + feat
    for (int t = 0; t < T_LEN; ++t) {
        const int cur = t & 1, nxt = cur ^ 1;
        if (tid < TAG) {
            float best = fv[cur][0] + tr[tid * TAG + 0];
            int bi = 0;
#pragma unroll
            for (int p = 1; p < TAG; ++p) {
                const float v = fv[cur][p] + tr[tid * TAG + p];
                if (v > best) { best = v; bi = p; }   // first-max on ties
            }
            bps[t * TAG + tid] = bi;
            fv[nxt][tid] = best + feats[t * TAG + tid];
        }
        __syncthreads();
    }

    if (tid == 0) {
        const int cur = T_LEN & 1;   // final fv buffer
        float best = fv[cur][0] + tr[STOP_TAG * TAG + 0];
        int last = 0;
#pragma unroll
        for (int p = 1; p < TAG; ++p) {
            const float v = fv[cur][p] + tr[STOP_TAG * TAG + p];
            if (v > best) { best = v; last = p; }
        }
        out[T_LEN] = best;           // score
        int tag = last;
        for (int t = T_LEN - 1; t >= 0; --t) {
            out[t] = (float)tag;
            tag = bps[t * TAG + tag];
        }
    }
}

// ---------------------------------------------------------------------------
// Launch
// ---------------------------------------------------------------------------
extern "C" void kernel_launch(void* const* d_in, const int* in_sizes, int n_in,
                              void* d_out, int out_size, void* d_ws, size_t ws_size,
                              hipStream_t stream) {
    const int*   sentence = (const int*)  d_in[0];
    const float* embed    = (const float*)d_in[1];
    const float* Wih_f    = (const float*)d_in[2];
    const float* Whh_f    = (const float*)d_in[3];
    const float* b_f      = (const float*)d_in[4];
    const float* Wih_b    = (const float*)d_in[5];
    const float* Whh_b    = (const float*)d_in[6];
    const float* b_b      = (const float*)d_in[7];
    const float* Wt       = (const float*)d_in[8];
    const float* bt       = (const float*)d_in[9];
    const float* trans    = (const float*)d_in[10];
    const float* h0       = (const float*)d_in[11];
    const float* c0       = (const float*)d_in[12];

    float* ws    = (float*)d_ws;
    float* Xp    = ws;                               // 4096*52
    float* Z     = Xp    + (size_t)T_LEN * KPAD;     // 4096*400
    float* hs_f  = Z     + (size_t)T_LEN * NFOUR;    // 4096*50
    float* hs_b  = hs_f  + (size_t)T_LEN * HID;      // 4096*50
    float* feats = hs_b  + (size_t)T_LEN * HID;      // 4096*5
    float* Bpt   = feats + (size_t)T_LEN * TAG;      // 400*52
    float* biasc = Bpt   + (size_t)NFOUR * KPAD;     // 400
    int*   bps   = (int*)(biasc + NFOUR);            // 4096*5 ints

    gather_pad_kernel<<<T_LEN, 64, 0, stream>>>(sentence, embed, Xp);
    pad_weights_kernel<<<NFOUR, 64, 0, stream>>>(Wih_f, Wih_b, b_f, b_b,
                                                 Bpt, biasc);

    dim3 gemm_grid(T_LEN / 16, NFOUR / 16);          // 256 x 25 tiles
    input_proj_wmma_kernel<<<gemm_grid, 32, 0, stream>>>(Xp, Bpt, biasc, Z);

    lstm_recurrence_kernel<<<2, 256, 0, stream>>>(
        Z, Whh_f, Whh_b, h0, c0, hs_f, hs_b);

    feats_viterbi_kernel<<<1, 256, 0, stream>>>(
        hs_f, hs_b, Wt, bt, trans, feats, bps, (float*)d_out);
}